// STGNN_91242285236402
// MI455X (gfx1250) — compile-verified
//
#include <hip/hip_runtime.h>

// Problem constants (B,T,N,F,H) = (8,16,64,16,128); E = N*(N-1) = 4032.
// Only t = T-1 contributes to the outputs (GRU result is discarded by the
// reference), so all stages run on a single time slice.
#define BB 8
#define TT 16
#define NN 64
#define FF 16
#define HH 128
#define EE 4032

#define NODE_PITCH 132   // 132 % 64 == 4 -> 16 consecutive rows hit distinct LDS banks
#define A_PITCH    68

typedef float v2f __attribute__((ext_vector_type(2)));
typedef float v8f __attribute__((ext_vector_type(8)));

// CDNA5 fp32 WMMA: D(16x16,f32) = A(16x4,f32) x B(4x16,f32) + C
// Fragment layout per ISA 7.12.2:
//   A: lane l holds row m = l&15; VGPR0/1 = K = 2*(l>>4) + {0,1}
//   B: lane l holds col n = l&15; VGPR0/1 = K = 2*(l>>4) + {0,1} (symmetric)
//   C: VGPR v holds row v + 8*(l>>4), col l&15
__device__ __forceinline__ v8f wmma4(v2f a, v2f b, v8f c) {
  return __builtin_amdgcn_wmma_f32_16x16x4_f32(
      /*neg_a=*/false, a, /*neg_b=*/false, b,
      /*c_mod=*/(short)0, c, /*reuse_a=*/false, /*reuse_b=*/false);
}

// ---------------------------------------------------------------------------
// Kernel 1 (one block per batch): node = X@Wg^T + bg  (LDS), then
// ps = node@W1s^T, pt = node@W1t^T + b1, lin = node@Wgcn^T + bgcn  -> d_ws
// ---------------------------------------------------------------------------
__global__ __launch_bounds__(256)
void stgnn_stage1(const float* __restrict__ x_seq,
                  const float* __restrict__ Wg,  const float* __restrict__ bg,
                  const float* __restrict__ W1,  const float* __restrict__ b1,
                  const float* __restrict__ Wgcn,const float* __restrict__ bgcn,
                  float* __restrict__ ws)
{
  __shared__ float node[NN * NODE_PITCH];

  const int b    = blockIdx.x;
  const int tid  = threadIdx.x;
  const int w    = tid >> 5;
  const int lane = tid & 31;
  const int ml   = lane & 15;        // A-row / B-col / C-col within tile
  const int kh   = (lane >> 4) * 2;  // K sub-offset for A/B fragments
  const int rsel = (lane >> 4) * 8;  // C row offset

  const float* X = x_seq + (size_t)((b * TT + (TT - 1)) * NN) * FF; // [64][16]

  // ---- Stage A: node (64x128) = X (64x16) @ Wg^T (16x128), 32 tiles / 8 waves
  for (int t4 = 0; t4 < 4; ++t4) {
    const int tile = w + 8 * t4;
    const int tm = tile >> 3, tn = tile & 7;
    const float bias = bg[tn * 16 + ml];
    v8f acc;
#pragma unroll
    for (int v = 0; v < 8; ++v) acc[v] = bias;
#pragma unroll
    for (int kk = 0; kk < FF; kk += 4) {
      v2f a, bf;
      a.x  = X[(tm * 16 + ml) * FF + kk + kh];
      a.y  = X[(tm * 16 + ml) * FF + kk + kh + 1];
      bf.x = Wg[(tn * 16 + ml) * FF + kk + kh];      // B[k][n] = Wg[h=n][f=k]
      bf.y = Wg[(tn * 16 + ml) * FF + kk + kh + 1];
      acc = wmma4(a, bf, acc);
    }
#pragma unroll
    for (int v = 0; v < 8; ++v)
      node[(tm * 16 + rsel + v) * NODE_PITCH + tn * 16 + ml] = acc[v];
  }
  __syncthreads();

  // ---- Stage B: 96 WMMA tile jobs (3 matrices x 32 tiles), K = 128
  float* wsb = ws + (size_t)b * 3 * NN * HH;   // ps | pt | lin
  for (int j4 = 0; j4 < 12; ++j4) {
    const int job  = w + 8 * j4;               // 0..95, unique per (wave,iter)
    const int mat  = job >> 5;                 // 0=ps, 1=pt, 2=lin
    const int tile = job & 31;
    const int tm = tile >> 3, tn = tile & 7;
    const int kout = tn * 16 + ml;             // output column 0..127

    const float* Bw;
    float bias;
    if (mat == 0)      { Bw = W1   + kout * (2 * HH);      bias = 0.0f;       }
    else if (mat == 1) { Bw = W1   + kout * (2 * HH) + HH; bias = b1[kout];   }
    else               { Bw = Wgcn + kout * HH;            bias = bgcn[kout]; }
    float* dst = wsb + (size_t)mat * NN * HH;

    v8f acc;
#pragma unroll
    for (int v = 0; v < 8; ++v) acc[v] = bias;
    for (int h0 = 0; h0 < HH; h0 += 4) {
      v2f a, bf;
      a.x  = node[(tm * 16 + ml) * NODE_PITCH + h0 + kh];
      a.y  = node[(tm * 16 + ml) * NODE_PITCH + h0 + kh + 1];
      bf.x = Bw[h0 + kh];
      bf.y = Bw[h0 + kh + 1];
      acc = wmma4(a, bf, acc);
    }
#pragma unroll
    for (int v = 0; v < 8; ++v)
      dst[(tm * 16 + rsel + v) * HH + tn * 16 + ml] = acc[v];
  }
}

// ---------------------------------------------------------------------------
// Kernel 2 (one block per batch):
//   A[s][t] = b2 + sum_k relu(ps[s,k]+pt[t,k]) * W2[k]   (A[s][s]=0), ew out
//   gcn     = A^T @ lin  (WMMA)
//   pressures[n] = gcn[n,:] . Wp + bp
// ---------------------------------------------------------------------------
__global__ __launch_bounds__(256)
void stgnn_stage2(const float* __restrict__ W2, const float* __restrict__ b2,
                  const float* __restrict__ Wp, const float* __restrict__ bp,
                  const float* __restrict__ ws, float* __restrict__ out)
{
  __shared__ float Asm[NN * A_PITCH];
  __shared__ float gcn[NN * NODE_PITCH];
  __shared__ float w2s[HH];
  __shared__ float wps[HH];

  const int b   = blockIdx.x;
  const int tid = threadIdx.x;

  if (tid < HH) { w2s[tid] = W2[tid]; wps[tid] = Wp[tid]; }
  __syncthreads();

  const float* wsb = ws + (size_t)b * 3 * NN * HH;
  const float* ps  = wsb;
  const float* pt  = wsb + NN * HH;
  const float* lin = wsb + 2 * NN * HH;
  const float  b2v = b2[0];

  // ---- Stage C: dense edge-weight matrix + ew[:, -1] output
  // 4096 (s,t) pairs over 256 threads; permutation edge order is analytic:
  // e = s*63 + (t<s ? t : t-1)
  for (int i = 0; i < 16; ++i) {
    const int idx = tid + 256 * i;
    const int s = idx >> 6, t = idx & 63;
    float val = 0.0f;
    if (s != t) {
      const float* prow = ps + s * HH;
      const float* qrow = pt + t * HH;
      float accv = b2v;
#pragma unroll 4
      for (int k = 0; k < HH; ++k)
        accv = fmaf(fmaxf(prow[k] + qrow[k], 0.0f), w2s[k], accv);
      val = accv;
      const int e_idx = s * 63 + ((t < s) ? t : t - 1);
      out[BB * NN + (size_t)b * EE + e_idx] = val;
    }
    Asm[s * A_PITCH + t] = val;
  }
  __syncthreads();

  // ---- Stage D: gcn (64x128) = A^T (64x64) @ lin (64x128) via WMMA, K=64
  const int w    = tid >> 5;
  const int lane = tid & 31;
  const int ml   = lane & 15;
  const int kh   = (lane >> 4) * 2;
  const int rsel = (lane >> 4) * 8;
  for (int j4 = 0; j4 < 4; ++j4) {
    const int tile = w + 8 * j4;
    const int tm = tile >> 3, tn = tile & 7;
    v8f acc;
#pragma unroll
    for (int v = 0; v < 8; ++v) acc[v] = 0.0f;
#pragma unroll 4
    for (int k0 = 0; k0 < NN; k0 += 4) {
      v2f a, bf;
      // A-operand element [m][k] = Asm[k][m]  (transposed read from LDS)
      a.x  = Asm[(k0 + kh)     * A_PITCH + tm * 16 + ml];
      a.y  = Asm[(k0 + kh + 1) * A_PITCH + tm * 16 + ml];
      bf.x = lin[(k0 + kh)     * HH + tn * 16 + ml];
      bf.y = lin[(k0 + kh + 1) * HH + tn * 16 + ml];
      acc = wmma4(a, bf, acc);
    }
#pragma unroll
    for (int v = 0; v < 8; ++v)
      gcn[(tm * 16 + rsel + v) * NODE_PITCH + tn * 16 + ml] = acc[v];
  }
  __syncthreads();

  // ---- Stage E: pressures
  if (tid < NN) {
    float accv = bp[0];
#pragma unroll 4
    for (int k = 0; k < HH; ++k)
      accv = fmaf(gcn[tid * NODE_PITCH + k], wps[k], accv);
    out[(size_t)b * NN + tid] = accv;
  }
}

extern "C" void kernel_launch(void* const* d_in, const int* in_sizes, int n_in,
                              void* d_out, int out_size, void* d_ws, size_t ws_size,
                              hipStream_t stream)
{
  (void)in_sizes; (void)n_in; (void)out_size; (void)ws_size;
  const float* x_seq = (const float*)d_in[0];
  // d_in[1] = edge_index: unused (all-permutations order is analytic)
  const float* Wg   = (const float*)d_in[2];
  const float* bg   = (const float*)d_in[3];
  const float* W1   = (const float*)d_in[4];
  const float* b1   = (const float*)d_in[5];
  const float* W2   = (const float*)d_in[6];
  const float* b2   = (const float*)d_in[7];
  const float* Wgcn = (const float*)d_in[8];
  const float* bgcn = (const float*)d_in[9];
  // d_in[10..13] = Wih/Whh/bih/bhh: dead code in the reference (GRU output unused)
  const float* Wp   = (const float*)d_in[14];
  const float* bp   = (const float*)d_in[15];

  float* ws  = (float*)d_ws;   // needs 8 * 3 * 64 * 128 * 4 = 786,432 bytes
  float* out = (float*)d_out;  // [512 pressures | 8*4032 ew_last]

  stgnn_stage1<<<BB, 256, 0, stream>>>(x_seq, Wg, bg, W1, b1, Wgcn, bgcn, ws);
  stgnn_stage2<<<BB, 256, 0, stream>>>(W2, b2, Wp, bp, ws, out);
}